// AddNoise_17824114278510
// MI455X (gfx1250) — compile-verified
//
#include <hip/hip_runtime.h>

typedef float v2f __attribute__((ext_vector_type(2)));
typedef float v8f __attribute__((ext_vector_type(8)));

// ---- workspace layout (float offsets) ----
#define U_OFF    ((size_t)0)           // [2][9][576][512]        = 5,308,416
#define Y1P_OFF  ((size_t)5308416)     // [2][171][36][512]       = 6,303,744
#define S_OFF    ((size_t)11612160)    // pair-packed [2304][352][2] = 1,622,016
#define Y2_OFF   ((size_t)13234176)    // [342][512]              =   175,104
#define W1P_OFF  ((size_t)13409280)    // [9][384][512][2]        = 3,538,944
#define W2P_OFF  ((size_t)16948224)    // [2304][512][2]          = 2,359,296
#define XP_OFF   ((size_t)19307520)    // [2][384][576][2]        =   884,736
// total = 20,192,256 floats = 80.8 MB

// noise (2,768,577) -> xp[b][ci/2][p][2]  (drop CLS token; K pair-packed)
__global__ void repack_xp_k(const float* __restrict__ noise, float* __restrict__ xp) {
    int idx = blockIdx.x * 256 + threadIdx.x;
    if (idx >= 2 * 768 * 576) return;
    int slot = idx & 1;
    int t = idx >> 1;
    int p = t % 576; t /= 576;
    int cp = t % 384; int b = t / 384;
    int ci = cp * 2 + slot;
    xp[idx] = noise[((size_t)b * 768 + ci) * 577 + 1 + p];
}

// w1 (512,768,3,3) -> w1p[tap][ci/2][co][2]
__global__ void repack_w1_k(const float* __restrict__ w1, float* __restrict__ w1p) {
    int idx = blockIdx.x * 256 + threadIdx.x;
    if (idx >= 9 * 768 * 512) return;
    int slot = idx & 1;
    int co = (idx >> 1) & 511;
    int t = (idx >> 1) >> 9;
    int cp = t % 384; int tap = t / 384;
    int ci = cp * 2 + slot;
    w1p[idx] = w1[((size_t)co * 768 + ci) * 9 + tap];
}

// w2 (512,512,3,3) -> w2p[K/2][co][2], K = ci*9+tap
__global__ void repack_w2_k(const float* __restrict__ w2, float* __restrict__ w2p) {
    int idx = blockIdx.x * 256 + threadIdx.x;
    if (idx >= 4608 * 512) return;
    int slot = idx & 1;
    int co = (idx >> 1) & 511;
    int kp = (idx >> 1) >> 9;
    int k = kp * 2 + slot;
    w2p[idx] = w2[(size_t)co * 4608 + k];
}

// GEMM1: u[b][tap][p][co] = sum_ci X[b][ci][p] * w1[co][ci][tap]
// one wave per 16x64 strip (4 co-tiles share one A fragment); K=768 via V_WMMA_F32_16X16X4_F32
__global__ void gemm1_k(const float* __restrict__ xp_f, const float* __restrict__ w1p_f,
                        float* __restrict__ u) {
    const v2f* Xp = (const v2f*)xp_f;            // [b][ci/2][p] of (K,K+1) pairs
    const v2f* Wp = (const v2f*)w1p_f;           // [tap][ci/2][co] of (K,K+1) pairs
    int id = blockIdx.x * 8 + (threadIdx.x >> 5);    // 5184 waves total
    int lane = threadIdx.x & 31;
    int lh = lane & 15;
    int half = lane >> 4;
    int cg = id & 7;                             // 8 co-groups of 64
    int t1 = id >> 3;
    int pt = t1 % 36;                            // 36 p-tiles
    int bt = t1 / 36;                            // 18 (b,tap)
    int b = bt / 9, tap = bt % 9;
    int p0 = pt * 16, co0 = cg * 64;
    const v2f* Xb = Xp + (size_t)b * 384 * 576 + p0 + lh;
    const v2f* Wb = Wp + (size_t)tap * 384 * 512 + co0 + lh;
    v8f acc[4];
#pragma unroll
    for (int j = 0; j < 4; j++)
#pragma unroll
        for (int i = 0; i < 8; i++) acc[j][i] = 0.0f;
    for (int k0 = 0; k0 < 768; k0 += 4) {
        int kp = (k0 >> 1) + half;               // lane-half selects K / K+2 pair (ISA layout)
        v2f a = Xb[(size_t)kp * 576];
#pragma unroll
        for (int j = 0; j < 4; j++) {
            v2f bm = Wb[(size_t)kp * 512 + j * 16];
            acc[j] = __builtin_amdgcn_wmma_f32_16x16x4_f32(false, a, false, bm, (short)0,
                                                           acc[j], false, false);
        }
    }
    float* U = u + (((size_t)(b * 9 + tap) * 576 + p0) * 512) + co0 + lh;
#pragma unroll
    for (int j = 0; j < 4; j++)
#pragma unroll
        for (int v = 0; v < 8; v++) {
            int m = v + half * 8;                // C layout: VGPR v -> rows v / v+8
            U[(size_t)m * 512 + j * 16] = acc[j][v];
        }
}

// y1p[b][cl][i*6+j][ci] = b1[ci]  (classes absent from the image stay at bias)
__global__ void init_y1p_k(const float* __restrict__ b1, float* __restrict__ y1p) {
    int idx = blockIdx.x * 256 + threadIdx.x;
    if (idx >= 2 * 171 * 36 * 512) return;
    y1p[idx] = b1[idx % 512];
}

// masked accumulate + 4x4 avg-pool: one workgroup per (b, 6x6 output block)
__global__ void mask_acc_k(const float* __restrict__ u, const int* __restrict__ targets,
                           float* __restrict__ y1p) {
    __shared__ int mcls[36];
    __shared__ int mok[36];
    int wid = blockIdx.x;
    int b = wid / 36; int blk = wid % 36;
    int bi = blk / 6, bj = blk % 6;
    int t = threadIdx.x;
    if (t < 36) {
        int dr = t / 6, dc = t % 6;
        int ph = 4 * bi - 1 + dr, pw = 4 * bj - 1 + dc;
        int ok = (ph >= 0 && ph < 24 && pw >= 0 && pw < 24);
        int c = 0;
        if (ok) c = targets[(size_t)b * 384 * 384 + (size_t)(ph * 16) * 384 + pw * 16];
        mok[t] = ok && (c >= 0) && (c < 171);
        mcls[t] = c;
    }
    __syncthreads();
    for (int e = 0; e < 36; e++) {               // uniform serial loop -> deterministic
        if (!mok[e]) continue;
        int dr = e / 6, dc = e % 6;
        int ph = 4 * bi - 1 + dr, pw = 4 * bj - 1 + dc;
        int plin = ph * 24 + pw;
        float s0 = 0.0f, s1 = 0.0f;
        for (int kh = 0; kh < 3; kh++) {
            if (kh > dr || kh < dr - 3) continue;     // tap lands inside this 4x4 block (rows)
            for (int kw = 0; kw < 3; kw++) {
                if (kw > dc || kw < dc - 3) continue; // (cols)
                const float* up = u + (((size_t)(b * 9 + kh * 3 + kw) * 576 + plin) * 512);
                s0 += up[t];
                s1 += up[t + 256];
            }
        }
        float* yp = y1p + (((size_t)(b * 171 + mcls[e]) * 36 + blk) * 512);
        yp[t]       += s0 * 0.0625f;
        yp[t + 256] += s1 * 0.0625f;
    }
}

// S (pair-packed [K/2][n][2]) : sub-rectangle sums of the 6x6 maps
// (conv2 SAME + global mean commuted); K = ci*9 + kh*3 + kw, n = b*171+cl (padded to 352)
__global__ void sum_S_k(const float* __restrict__ y1p, float* __restrict__ S) {
    int idx = blockIdx.x * 256 + threadIdx.x;
    if (idx >= 342 * 512) return;
    int ci = idx % 512; int n = idx / 512;
    const float* v = y1p + (size_t)n * 36 * 512 + ci;
    float r0[6], r1[6], r2[6];
#pragma unroll
    for (int i = 0; i < 6; i++) {
        float a0 = v[(size_t)(i * 6 + 0) * 512];
        float a1 = v[(size_t)(i * 6 + 1) * 512];
        float a2 = v[(size_t)(i * 6 + 2) * 512];
        float a3 = v[(size_t)(i * 6 + 3) * 512];
        float a4 = v[(size_t)(i * 6 + 4) * 512];
        float a5 = v[(size_t)(i * 6 + 5) * 512];
        r1[i] = a0 + a1 + a2 + a3 + a4 + a5;     // kw=1: cols 0..5
        r0[i] = r1[i] - a5;                      // kw=0: cols 0..4
        r2[i] = r1[i] - a0;                      // kw=2: cols 1..5
    }
    float c0 = 0, c1 = 0, c2 = 0;
#pragma unroll
    for (int i = 0; i < 6; i++) { c0 += r0[i]; c1 += r1[i]; c2 += r2[i]; }
    float vals[9];
    vals[0] = c0 - r0[5]; vals[1] = c1 - r1[5]; vals[2] = c2 - r2[5]; // kh=0: rows 0..4
    vals[3] = c0;         vals[4] = c1;         vals[5] = c2;         // kh=1: rows 0..5
    vals[6] = c0 - r0[0]; vals[7] = c1 - r1[0]; vals[8] = c2 - r2[0]; // kh=2: rows 1..5
#pragma unroll
    for (int kk = 0; kk < 9; kk++) {
        int k = ci * 9 + kk;
        S[((size_t)(k >> 1) * 352 + n) * 2 + (k & 1)] = vals[kk];
    }
}

// GEMM2: y2[n][co] = b2[co] + (1/36) * sum_K S[K][n] * w2[co][K]
// M=n(342 pad 352), N=co(512), K=4608; 4 co-tiles per wave
__global__ void gemm2_k(const float* __restrict__ S_f, const float* __restrict__ w2p_f,
                        const float* __restrict__ b2, float* __restrict__ y2) {
    const v2f* Sp = (const v2f*)S_f;             // [K/2][n] of (K,K+1) pairs
    const v2f* Wp = (const v2f*)w2p_f;           // [K/2][co] of (K,K+1) pairs
    int id = blockIdx.x * 8 + (threadIdx.x >> 5);    // 176 waves total
    int lane = threadIdx.x & 31;
    int lh = lane & 15;
    int half = lane >> 4;
    int cg = id & 7;                             // 8 co-groups of 64
    int nt = id >> 3;                            // 22 n-tiles
    int n0 = nt * 16, co0 = cg * 64;
    bool arow = (n0 + lh) < 342;
    v8f acc[4];
#pragma unroll
    for (int j = 0; j < 4; j++)
#pragma unroll
        for (int i = 0; i < 8; i++) acc[j][i] = 0.0f;
    for (int k0 = 0; k0 < 4608; k0 += 4) {
        int kp = (k0 >> 1) + half;
        v2f a = {0.0f, 0.0f};
        if (arow) a = Sp[(size_t)kp * 352 + n0 + lh];
#pragma unroll
        for (int j = 0; j < 4; j++) {
            v2f bm = Wp[(size_t)kp * 512 + co0 + j * 16 + lh];
            acc[j] = __builtin_amdgcn_wmma_f32_16x16x4_f32(false, a, false, bm, (short)0,
                                                           acc[j], false, false);
        }
    }
#pragma unroll
    for (int j = 0; j < 4; j++) {
        float bias = b2[co0 + j * 16 + lh];
#pragma unroll
        for (int v = 0; v < 8; v++) {
            int m = n0 + v + half * 8;
            if (m < 342)
                y2[(size_t)m * 512 + co0 + j * 16 + lh] = bias + acc[j][v] * (1.0f / 36.0f);
        }
    }
}

// outputs: image_noise = transpose(noise,(2,0,1)); text_noise = w3[k]*y2 + b3[k]
__global__ void finalize_k(const float* __restrict__ noise, const float* __restrict__ y2,
                           const float* __restrict__ w3, const float* __restrict__ b3,
                           float* __restrict__ out) {
    size_t idx = (size_t)blockIdx.x * 256 + threadIdx.x;
    if (idx < 886272) {
        int p = (int)(idx / 1536); int r = (int)(idx % 1536);
        int b = r / 768; int ci = r % 768;
        out[idx] = noise[((size_t)b * 768 + ci) * 577 + p];
    } else if (idx < 14369280) {
        size_t j = idx - 886272;
        int co = (int)(j % 512); size_t t = j / 512;
        int cl = (int)(t % 171); size_t t2 = t / 171;
        int k = (int)(t2 % 77); int b = (int)(t2 / 77);
        out[idx] = w3[k] * y2[((size_t)(b * 171 + cl)) * 512 + co] + b3[k];
    }
}

extern "C" void kernel_launch(void* const* d_in, const int* in_sizes, int n_in,
                              void* d_out, int out_size, void* d_ws, size_t ws_size,
                              hipStream_t stream) {
    const float* noise   = (const float*)d_in[0];
    const int*   targets = (const int*)  d_in[1];
    const float* w1      = (const float*)d_in[2];
    const float* b1      = (const float*)d_in[3];
    const float* w2      = (const float*)d_in[4];
    const float* b2      = (const float*)d_in[5];
    const float* w3      = (const float*)d_in[6];
    const float* b3      = (const float*)d_in[7];
    float* out = (float*)d_out;
    float* ws  = (float*)d_ws;

    float* u   = ws + U_OFF;
    float* y1p = ws + Y1P_OFF;
    float* S   = ws + S_OFF;
    float* y2  = ws + Y2_OFF;
    float* w1p = ws + W1P_OFF;
    float* w2p = ws + W2P_OFF;
    float* xp  = ws + XP_OFF;

    repack_xp_k<<<3456, 256, 0, stream>>>(noise, xp);
    repack_w1_k<<<13824, 256, 0, stream>>>(w1, w1p);
    repack_w2_k<<<9216, 256, 0, stream>>>(w2, w2p);
    gemm1_k<<<648, 256, 0, stream>>>(xp, w1p, u);
    init_y1p_k<<<24624, 256, 0, stream>>>(b1, y1p);
    mask_acc_k<<<72, 256, 0, stream>>>(u, targets, y1p);
    sum_S_k<<<684, 256, 0, stream>>>(y1p, S);
    gemm2_k<<<22, 256, 0, stream>>>(S, w2p, b2, y2);
    finalize_k<<<56130, 256, 0, stream>>>(noise, y2, w3, b3, out);
}